// ExternalNeighbors_61787399520639
// MI455X (gfx1250) — compile-verified
//
#include <hip/hip_runtime.h>
#include <hip/hip_bf16.h>

typedef __attribute__((ext_vector_type(4))) float floatx4;

#define TPB   256
#define PPT   4
#define TILE  (TPB * PPT)            // 1024 pairs per block-tile
#define HARD_CUT 2.0f

// ---- CDNA5 async copy: global -> LDS, 16B per lane, non-temporal ----------
__device__ __forceinline__ void async_ld16(unsigned lds_off, const void* g) {
  asm volatile("global_load_async_to_lds_b128 %0, %1, off th:TH_LOAD_NT"
               :: "v"(lds_off), "v"(g) : "memory");
}
__device__ __forceinline__ void wait_async_le5() {
  asm volatile("s_wait_asynccnt 0x5" ::: "memory");
}
__device__ __forceinline__ void wait_async_0() {
  asm volatile("s_wait_asynccnt 0x0" ::: "memory");
}

__global__ __launch_bounds__(TPB)
void neigh_main(const float* __restrict__ coords,      // [n_real*3] (L2-resident)
                const int*   __restrict__ real_atoms,  // [n_real]   (L2-resident)
                const int*   __restrict__ shifts,      // [n*3]      (streamed)
                const float* __restrict__ cell,        // [9] uniform
                const int*   __restrict__ pair_first,  // [n] (streamed)
                const int*   __restrict__ pair_second, // [n] (streamed)
                float* __restrict__ out,
                long long nll, int tiles)
{
  __shared__ int s_pf[2][TILE];
  __shared__ int s_ps[2][TILE];
  __shared__ int s_sh[2][TILE * 3];

  const int tid = threadIdx.x;
  const size_t n = (size_t)nll;

  // Uniform 3x3 cell -> scalar loads
  const float c00 = cell[0], c01 = cell[1], c02 = cell[2];
  const float c10 = cell[3], c11 = cell[4], c12 = cell[5];
  const float c20 = cell[6], c21 = cell[7], c22 = cell[8];

  auto issue = [&](int t, int buf) {
    const size_t p0 = (size_t)t * TILE;
    async_ld16((unsigned)(size_t)&s_pf[buf][tid * 4], pair_first  + p0 + tid * 4);
    async_ld16((unsigned)(size_t)&s_ps[buf][tid * 4], pair_second + p0 + tid * 4);
    const int* sh = shifts + p0 * 3 + tid * 12;       // 48B per thread, 16B aligned
    const unsigned lo = (unsigned)(size_t)&s_sh[buf][tid * 12];
    async_ld16(lo,      sh);
    async_ld16(lo + 16, sh + 4);
    async_ld16(lo + 32, sh + 8);
  };

  int t   = blockIdx.x;
  int buf = 0;
  if (t < tiles) issue(t, 0);

  while (t < tiles) {
    const int tn = t + gridDim.x;
    if (tn < tiles) {
      issue(tn, buf ^ 1);        // next tile in flight (5 async ops)
      wait_async_le5();          // oldest 5 (current tile) complete, in order
    } else {
      wait_async_0();
    }

    const size_t base = (size_t)t * TILE + (size_t)tid * PPT;
    floatx4 dv, pfv, psv, mv;
    float pcf[12];

#pragma unroll
    for (int j = 0; j < PPT; ++j) {
      const int pf = s_pf[buf][tid * 4 + j];
      const int ps = s_ps[buf][tid * 4 + j];
      const int a1 = real_atoms[pf];                 // L2-hit gathers
      const int a2 = real_atoms[ps];
      const float x1 = coords[a1 * 3 + 0], y1 = coords[a1 * 3 + 1], z1 = coords[a1 * 3 + 2];
      const float x2 = coords[a2 * 3 + 0], y2 = coords[a2 * 3 + 1], z2 = coords[a2 * 3 + 2];
      const float fx = (float)s_sh[buf][tid * 12 + j * 3 + 0];
      const float fy = (float)s_sh[buf][tid * 12 + j * 3 + 1];
      const float fz = (float)s_sh[buf][tid * 12 + j * 3 + 2];
      const float px = x2 - x1 + fx * c00 + fy * c10 + fz * c20;
      const float py = y2 - y1 + fx * c01 + fy * c11 + fz * c21;
      const float pz = z2 - z1 + fx * c02 + fy * c12 + fz * c22;
      const float d  = sqrtf(px * px + py * py + pz * pz);
      const bool  m  = d < HARD_CUT;
      dv[j]  = m ? d          : 0.0f;
      pfv[j] = m ? (float)pf  : -1.0f;
      psv[j] = m ? (float)ps  : -1.0f;
      mv[j]  = m ? 1.0f       : 0.0f;
      pcf[3 * j + 0] = m ? px : 0.0f;
      pcf[3 * j + 1] = m ? py : 0.0f;
      pcf[3 * j + 2] = m ? pz : 0.0f;
    }
    const floatx4 pc0 = {pcf[0], pcf[1], pcf[2],  pcf[3]};
    const floatx4 pc1 = {pcf[4], pcf[5], pcf[6],  pcf[7]};
    const floatx4 pc2 = {pcf[8], pcf[9], pcf[10], pcf[11]};

    // Streaming non-temporal B128 stores (outputs exceed L2; keep L2 for gathers)
    __builtin_nontemporal_store(dv,  (floatx4*)(out +             base));
    __builtin_nontemporal_store(pfv, (floatx4*)(out +     n     + base));
    __builtin_nontemporal_store(psv, (floatx4*)(out + 2 * n     + base));
    __builtin_nontemporal_store(pc0, (floatx4*)(out + 3 * n + 3 * base));
    __builtin_nontemporal_store(pc1, (floatx4*)(out + 3 * n + 3 * base + 4));
    __builtin_nontemporal_store(pc2, (floatx4*)(out + 3 * n + 3 * base + 8));
    __builtin_nontemporal_store(mv,  (floatx4*)(out + 6 * n     + base));

    buf ^= 1;
    t = tn;
  }
}

// Scalar tail for n not divisible by TILE (direct loads, same math)
__global__ void neigh_tail(const float* __restrict__ coords,
                           const int*   __restrict__ real_atoms,
                           const int*   __restrict__ shifts,
                           const float* __restrict__ cell,
                           const int*   __restrict__ pair_first,
                           const int*   __restrict__ pair_second,
                           float* __restrict__ out,
                           long long nll, long long startll)
{
  const size_t n = (size_t)nll;
  const size_t i = (size_t)startll + (size_t)blockIdx.x * blockDim.x + threadIdx.x;
  if (i >= n) return;
  const int pf = pair_first[i], ps = pair_second[i];
  const int a1 = real_atoms[pf], a2 = real_atoms[ps];
  const float fx = (float)shifts[i * 3 + 0];
  const float fy = (float)shifts[i * 3 + 1];
  const float fz = (float)shifts[i * 3 + 2];
  const float px = coords[a2 * 3 + 0] - coords[a1 * 3 + 0] + fx * cell[0] + fy * cell[3] + fz * cell[6];
  const float py = coords[a2 * 3 + 1] - coords[a1 * 3 + 1] + fx * cell[1] + fy * cell[4] + fz * cell[7];
  const float pz = coords[a2 * 3 + 2] - coords[a1 * 3 + 2] + fx * cell[2] + fy * cell[5] + fz * cell[8];
  const float d = sqrtf(px * px + py * py + pz * pz);
  const bool  m = d < HARD_CUT;
  out[i]                 = m ? d         : 0.0f;
  out[n + i]             = m ? (float)pf : -1.0f;
  out[2 * n + i]         = m ? (float)ps : -1.0f;
  out[3 * n + 3 * i + 0] = m ? px : 0.0f;
  out[3 * n + 3 * i + 1] = m ? py : 0.0f;
  out[3 * n + 3 * i + 2] = m ? pz : 0.0f;
  out[6 * n + i]         = m ? 1.0f : 0.0f;
}

// Zero any d_out tail beyond the 7*n floats we define (harness poisons with 0xAA)
__global__ void zero_tail(float* __restrict__ out, long long start, long long end)
{
  const long long i = start + (long long)blockIdx.x * blockDim.x + threadIdx.x;
  if (i < end) out[i] = 0.0f;
}

extern "C" void kernel_launch(void* const* d_in, const int* in_sizes, int n_in,
                              void* d_out, int out_size, void* d_ws, size_t ws_size,
                              hipStream_t stream) {
  const float* coords      = (const float*)d_in[0];
  const int*   real_atoms  = (const int*)  d_in[1];
  const int*   shifts      = (const int*)  d_in[2];
  const float* cell        = (const float*)d_in[3];
  const int*   pair_first  = (const int*)  d_in[4];
  const int*   pair_second = (const int*)  d_in[5];
  float* out = (float*)d_out;

  const long long n     = (long long)in_sizes[4];   // 8388608
  const int       tiles = (int)(n / TILE);          // 8192
  const long long start = (long long)tiles * TILE;

  if (tiles > 0) {
    const int grid = tiles < 2048 ? tiles : 2048;   // 4 tiles per block at full size
    neigh_main<<<grid, TPB, 0, stream>>>(coords, real_atoms, shifts, cell,
                                         pair_first, pair_second, out, n, tiles);
  }
  const long long rem = n - start;
  if (rem > 0) {
    neigh_tail<<<(int)((rem + 255) / 256), 256, 0, stream>>>(
        coords, real_atoms, shifts, cell, pair_first, pair_second, out, n, start);
  }
  const long long defined = 7 * n;
  if ((long long)out_size > defined) {
    const long long extra = (long long)out_size - defined;
    zero_tail<<<(int)((extra + 255) / 256), 256, 0, stream>>>(out, defined, (long long)out_size);
  }
}